// basic_gcn_53214644797577
// MI455X (gfx1250) — compile-verified
//
#include <hip/hip_runtime.h>

// GCN on MI455X (gfx1250, wave32):
//   - dense transforms via V_WMMA_F32_16X16X4_F32 (fp32 in/out, 16x16 tile per wave),
//     K fully unrolled at compile time -> straight-line wmma stream
//   - edge aggregation via relaxed agent-scope float atomics (resolve in 192MB L2;
//     node feature table ~25.6MB fits entirely in L2, so gathers hit on-chip)
//   - self-loop + bias fused into the aggregation-init kernel

typedef __attribute__((ext_vector_type(2))) float v2f;
typedef __attribute__((ext_vector_type(8))) float v8f;

__device__ __forceinline__ void atomic_add_f32(float* p, float v) {
    __hip_atomic_fetch_add(p, v, __ATOMIC_RELAXED, __HIP_MEMORY_SCOPE_AGENT);
}

// ---------------------------------------------------------------------------
// Y[M,N] = (RELU ? relu(X) : X) @ W (+ bias if BIAS), one 16x16 tile per wave32.
// K fully unrolled in steps of 4 using V_WMMA_F32_16X16X4_F32.
//   A 16x4 f32: lane l -> row M=l%16, holds K = koff..koff+1 (koff = (l<16?0:2))
//   B 4x16 f32: lane l -> col N=l%16, same K split
//   C/D 16x16:  lane l -> col N=l%16, VGPR r -> row M = r + (l<16?0:8)
// ---------------------------------------------------------------------------
template <int K, bool RELU, bool BIAS>
__global__ __launch_bounds__(32)
void wmma_gemm_f32(const float* __restrict__ X, int ldx,
                   const float* __restrict__ W, int ldw,
                   float* __restrict__ Y, int ldy,
                   const float* __restrict__ bias)
{
    const int lane = threadIdx.x & 31;
    const int half = lane >> 4;       // 0: lanes 0-15, 1: lanes 16-31
    const int l16  = lane & 15;
    const size_t m = (size_t)blockIdx.x * 16 + l16;   // A row for this lane
    const int    n = blockIdx.y * 16 + l16;           // B/D column for this lane
    const int koff = half * 2;

    v8f c = {};
#pragma unroll
    for (int k0 = 0; k0 < K; k0 += 4) {
        const float* ap = X + m * (size_t)ldx + (k0 + koff);
        v2f a;
        a.x = ap[0];
        a.y = ap[1];
        if (RELU) {
            a.x = fmaxf(a.x, 0.0f);
            a.y = fmaxf(a.y, 0.0f);
        }
        v2f b;
        b.x = W[(size_t)(k0 + koff + 0) * ldw + n];
        b.y = W[(size_t)(k0 + koff + 1) * ldw + n];
        // 8 args: (neg_a, A, neg_b, B, c_mod, C, reuse_a, reuse_b)
        c = __builtin_amdgcn_wmma_f32_16x16x4_f32(false, a, false, b,
                                                  (short)0, c, false, false);
    }

    const float bv = BIAS ? bias[n] : 0.0f;
    const size_t mbase = (size_t)blockIdx.x * 16 + half * 8;
#pragma unroll
    for (int r = 0; r < 8; ++r) {
        Y[(mbase + r) * (size_t)ldy + n] = c[r] + bv;
    }
}

// ---------------------------------------------------------------------------
// Degree / normalization
// ---------------------------------------------------------------------------
__global__ __launch_bounds__(256)
void deg_init(float* __restrict__ deg, int N) {
    int i = blockIdx.x * blockDim.x + threadIdx.x;
    if (i < N) deg[i] = 1.0f;   // self-loop weight
}

__global__ __launch_bounds__(256)
void deg_accum(const int* __restrict__ ei, const float* __restrict__ ew,
               float* __restrict__ deg, int E) {
    int e = blockIdx.x * blockDim.x + threadIdx.x;
    if (e < E) atomic_add_f32(&deg[ei[E + e]], ew[e]);   // dst row of edge_index
}

__global__ __launch_bounds__(256)
void deg_to_dinv(float* __restrict__ deg, int N) {
    int i = blockIdx.x * blockDim.x + threadIdx.x;
    if (i < N) {
        float v = deg[i];
        deg[i] = (v > 0.0f) ? rsqrtf(v) : 0.0f;
    }
}

// ---------------------------------------------------------------------------
// out[i,f] = bias[f] + t[i,f] * dinv[i]^2      (self-loop term + bias)
// FSHIFT = log2(F)
// ---------------------------------------------------------------------------
template <int FSHIFT>
__global__ __launch_bounds__(256)
void agg_init(const float* __restrict__ T, const float* __restrict__ dinv,
              const float* __restrict__ bias, float* __restrict__ out,
              long long total)
{
    long long tid = (long long)blockIdx.x * blockDim.x + threadIdx.x;
    if (tid >= total) return;
    int i = (int)(tid >> FSHIFT);
    int f = (int)tid & ((1 << FSHIFT) - 1);
    float di = dinv[i];
    out[tid] = bias[f] + T[tid] * di * di;
}

// ---------------------------------------------------------------------------
// Per edge e: out[dst] += t[src] * dinv[src]*ew[e]*dinv[dst]
// One thread handles 4 features (float4 gather + 4 atomics).
// F = feature width (compile-time), CSHIFT = log2(F/4).
// ---------------------------------------------------------------------------
template <int F, int CSHIFT>
__global__ __launch_bounds__(256)
void edge_scatter(const float* __restrict__ T, const int* __restrict__ ei,
                  const float* __restrict__ ew, const float* __restrict__ dinv,
                  float* __restrict__ out, int E)
{
    long long tid = (long long)blockIdx.x * blockDim.x + threadIdx.x;
    int e = (int)(tid >> CSHIFT);
    if (e >= E) return;
    int c4 = ((int)tid & ((1 << CSHIFT) - 1)) << 2;

    int s = ei[e];          // src row
    int d = ei[E + e];      // dst row
    float norm = dinv[s] * ew[e] * dinv[d];

    const float4 v = *(const float4*)(T + (size_t)s * F + c4);
    float* o = out + (size_t)d * F + c4;
    atomic_add_f32(o + 0, v.x * norm);
    atomic_add_f32(o + 1, v.y * norm);
    atomic_add_f32(o + 2, v.z * norm);
    atomic_add_f32(o + 3, v.w * norm);
}

// ---------------------------------------------------------------------------
extern "C" void kernel_launch(void* const* d_in, const int* in_sizes, int n_in,
                              void* d_out, int out_size, void* d_ws, size_t ws_size,
                              hipStream_t stream)
{
    const float* x  = (const float*)d_in[0];
    const int*   ei = (const int*)  d_in[1];   // [2,E] row-major: [0:E)=src, [E:2E)=dst
    const float* ew = (const float*)d_in[2];
    const float* W1 = (const float*)d_in[3];
    const float* b1 = (const float*)d_in[4];
    const float* W2 = (const float*)d_in[5];
    const float* b2 = (const float*)d_in[6];
    const float* W3 = (const float*)d_in[7];
    const float* b3 = (const float*)d_in[8];
    const float* Wf = (const float*)d_in[9];
    const float* bf = (const float*)d_in[10];
    float* out = (float*)d_out;

    const int N = in_sizes[0] / 64;       // 100000 (multiple of 16)
    const int E = in_sizes[1] / 2;        // 1600000

    // workspace: dinv [N] | tmp [N*64] | feat [N*64]   (~52 MB)
    char* ws = (char*)d_ws;
    float* dinv = (float*)ws;
    size_t off  = (((size_t)N * 4) + 255) & ~(size_t)255;
    float* tmp  = (float*)(ws + off);
    float* feat = (float*)(ws + off + (size_t)N * 64 * sizeof(float));

    const long long tot64 = (long long)N * 64;
    const long long tot32 = (long long)N * 32;
    const long long esc64 = (long long)E * 16;   // E * (64/4)
    const long long esc32 = (long long)E * 8;    // E * (32/4)

    // --- normalization: deg -> dinv (in place) ---
    deg_init   <<<(N + 255) / 256, 256, 0, stream>>>(dinv, N);
    deg_accum  <<<(E + 255) / 256, 256, 0, stream>>>(ei, ew, dinv, E);
    deg_to_dinv<<<(N + 255) / 256, 256, 0, stream>>>(dinv, N);

    dim3 g64(N / 16, 64 / 16);   // 6250 x 4 tiles
    dim3 g32(N / 16, 32 / 16);   // 6250 x 2 tiles

    // --- layer 1: tmp = x@W1 ; feat = b1 + self + scatter
    wmma_gemm_f32<64, false, false><<<g64, 32, 0, stream>>>(x, 64, W1, 64, tmp, 64, nullptr);
    agg_init<6>      <<<(tot64 + 255) / 256, 256, 0, stream>>>(tmp, dinv, b1, feat, tot64);
    edge_scatter<64, 4><<<(esc64 + 255) / 256, 256, 0, stream>>>(tmp, ei, ew, dinv, feat, E);

    // --- layer 2: tmp = relu(feat)@W2 ; feat = b2 + self + scatter
    wmma_gemm_f32<64, true, false><<<g64, 32, 0, stream>>>(feat, 64, W2, 64, tmp, 64, nullptr);
    agg_init<6>      <<<(tot64 + 255) / 256, 256, 0, stream>>>(tmp, dinv, b2, feat, tot64);
    edge_scatter<64, 4><<<(esc64 + 255) / 256, 256, 0, stream>>>(tmp, ei, ew, dinv, feat, E);

    // --- layer 3 (OUT=32): tmp = relu(feat)@W3 ; feat (as [N,32], stride 32) = b3 + self + scatter
    wmma_gemm_f32<64, true, false><<<g32, 32, 0, stream>>>(feat, 64, W3, 32, tmp, 32, nullptr);
    agg_init<5>      <<<(tot32 + 255) / 256, 256, 0, stream>>>(tmp, dinv, b3, feat, tot32);
    edge_scatter<32, 3><<<(esc32 + 255) / 256, 256, 0, stream>>>(tmp, ei, ew, dinv, feat, E);

    // --- final linear: out = feat @ Wf + bf
    wmma_gemm_f32<32, false, true><<<g32, 32, 0, stream>>>(feat, 32, Wf, 32, out, 32, bf);
}